// VectorQuantizer_79199196938798
// MI455X (gfx1250) — compile-verified
//
#include <hip/hip_runtime.h>

#define N_ROWS 65536   // 64*1024 latent vectors
#define NE     8192    // codebook entries
#define DDIM   256     // embedding dim
constexpr float BETA = 0.25f;

typedef __attribute__((ext_vector_type(2))) float        v2f;
typedef __attribute__((ext_vector_type(8))) float        v8f;
typedef __attribute__((ext_vector_type(4))) unsigned int u32x4;
typedef __attribute__((ext_vector_type(4))) int          i32x4;
typedef __attribute__((ext_vector_type(8))) int          i32x8;

// ---------------------------------------------------------------------------
// monotone key: float -> uint such that key order == float order (asc)
__device__ __forceinline__ unsigned int fkey(float f) {
    unsigned int b = __float_as_uint(f);
    return (b & 0x80000000u) ? ~b : (b | 0x80000000u);
}

// ---------------------------------------------------------------------------
// TDM: async 2D tile load Global->LDS.  tile = 64 (K, innermost) x 128 rows,
// tensor_dim0 = 256 (row length), tensor_dim0_stride = 256, data_size = 4B.
__device__ __forceinline__ void tdm_load_tile(unsigned int lds_off,
                                              unsigned long long gaddr,
                                              unsigned int tensor_dim1) {
    u32x4 g0;
    g0.x = 1u;                                                   // count=1 (valid), user mode
    g0.y = lds_off;                                              // lds_addr (bytes)
    g0.z = (unsigned int)gaddr;                                  // global_addr[31:0]
    g0.w = ((unsigned int)(gaddr >> 32) & 0x01FFFFFFu)           // global_addr[56:32]
         | (2u << 30);                                           // type = 2 ("image")
    i32x8 g1;
    g1[0] = (int)(2u << 16);                                     // wg_mask=0, data_size=4B
    g1[1] = (int)((DDIM & 0xFFFFu) << 16);                       // tensor_dim0 lo16 @ bits 63:48
    g1[2] = (int)((tensor_dim1 & 0xFFFFu) << 16);                // dim0 hi16=0 | tensor_dim1 lo16
    g1[3] = (int)((tensor_dim1 >> 16) | (64u << 16));            // tensor_dim1 hi16 | tile_dim0=64
    g1[4] = (int)128;                                            // tile_dim1=128, tile_dim2=0
    g1[5] = (int)DDIM;                                           // tensor_dim0_stride lo32
    g1[6] = 0;                                                   // stride hi / dim1_stride lo
    g1[7] = 0;
    i32x4 z4 = {0, 0, 0, 0};
#if defined(__clang_major__) && (__clang_major__ >= 23)
    i32x8 z8 = {0, 0, 0, 0, 0, 0, 0, 0};
    __builtin_amdgcn_tensor_load_to_lds(g0, g1, z4, z4, z8, 0);
#else
    __builtin_amdgcn_tensor_load_to_lds(g0, g1, z4, z4, 0);
#endif
}

// ---------------------------------------------------------------------------
__global__ void vq_init(unsigned long long* __restrict__ minpack) {
    int i = blockIdx.x * blockDim.x + threadIdx.x;
    if (i < N_ROWS) minpack[i] = ~0ull;
}

// ---------------------------------------------------------------------------
// one wave (32 lanes) per row: sum of squares over 256 floats
__global__ void vq_norms(const float* __restrict__ x, const float* __restrict__ emb,
                         float* __restrict__ zn, float* __restrict__ en) {
    int wave = (int)((blockIdx.x * blockDim.x + threadIdx.x) >> 5);
    int lane = threadIdx.x & 31;
    const float* src;
    float* dst;
    if (wave < N_ROWS) {
        src = x + (size_t)wave * DDIM;
        dst = zn + wave;
    } else if (wave < N_ROWS + NE) {
        src = emb + (size_t)(wave - N_ROWS) * DDIM;
        dst = en + (wave - N_ROWS);
    } else {
        return;
    }
    float4 p0 = *(const float4*)(src + lane * 4);
    float4 p1 = *(const float4*)(src + 128 + lane * 4);
    float s = p0.x * p0.x + p0.y * p0.y + p0.z * p0.z + p0.w * p0.w
            + p1.x * p1.x + p1.y * p1.y + p1.z * p1.z + p1.w * p1.w;
#pragma unroll
    for (int m = 16; m >= 1; m >>= 1) s += __shfl_xor(s, m, 32);
    if (lane == 0) *dst = s;
}

// ---------------------------------------------------------------------------
// 128x128 tile of d per block; 8 waves, each wave: 4 (M) x 2 (N) WMMA tiles.
// K=256 chunked by 64; chunks staged to double-buffered LDS by the Tensor
// Data Mover, software-pipelined so DMA of chunk k+1 overlaps WMMA on chunk k.
__global__ __launch_bounds__(256, 2) void vq_gemm(
        const float* __restrict__ x, const float* __restrict__ emb,
        const float* __restrict__ zn, const float* __restrict__ en,
        float* __restrict__ dOut, unsigned long long* __restrict__ minpack) {
    __shared__ float ldsA[2][128][64];   // latent rows  [buf][m][k]
    __shared__ float ldsB[2][128][64];   // emb rows     [buf][n][k]

    const int i0   = blockIdx.y * 128;
    const int j0   = blockIdx.x * 128;
    const int tid  = threadIdx.x;
    const int wave = tid >> 5;
    const int lane = tid & 31;
    const int lan  = lane & 15;
    const int hi   = lane >> 4;
    const int wm   = wave & 1;    // 2 row strips of 64
    const int wn   = wave >> 1;   // 4 col strips of 32

    v8f acc[4][2];
    {
        v8f zv = {0.f, 0.f, 0.f, 0.f, 0.f, 0.f, 0.f, 0.f};
#pragma unroll
        for (int r = 0; r < 4; ++r)
#pragma unroll
            for (int c = 0; c < 2; ++c) acc[r][c] = zv;
    }

    const unsigned long long xbase =
        (unsigned long long)(size_t)x + (unsigned long long)i0 * DDIM * 4ull;
    const unsigned long long ebase =
        (unsigned long long)(size_t)emb + (unsigned long long)j0 * DDIM * 4ull;

    // prologue: fill buffer 0 with K-chunk 0
    if (wave == 0) {
        tdm_load_tile((unsigned int)(size_t)(&ldsA[0][0][0]), xbase, (unsigned int)N_ROWS);
        tdm_load_tile((unsigned int)(size_t)(&ldsB[0][0][0]), ebase, (unsigned int)NE);
    }

#pragma unroll 1
    for (int kc = 0; kc < DDIM / 64; ++kc) {
        const int cur = kc & 1;
        if (wave == 0) __builtin_amdgcn_s_wait_tensorcnt(0);  // chunk kc landed in LDS
        __syncthreads();

        // prefetch chunk kc+1 into the alternate buffer (overlaps with WMMA below)
        if (wave == 0 && kc + 1 < DDIM / 64) {
            unsigned long long koff = (unsigned long long)(kc + 1) * 64ull * 4ull;
            tdm_load_tile((unsigned int)(size_t)(&ldsA[cur ^ 1][0][0]),
                          xbase + koff, (unsigned int)N_ROWS);
            tdm_load_tile((unsigned int)(size_t)(&ldsB[cur ^ 1][0][0]),
                          ebase + koff, (unsigned int)NE);
        }

#pragma unroll
        for (int k = 0; k < 64; k += 4) {
            const int kk = k + 2 * hi;    // f32 frag layout: lanes 0-15 K=k,k+1; 16-31 K=k+2,k+3
            v2f a[4], b[2];
#pragma unroll
            for (int r = 0; r < 4; ++r)
                a[r] = *(const v2f*)(&ldsA[cur][wm * 64 + r * 16 + lan][kk]);
#pragma unroll
            for (int c = 0; c < 2; ++c)
                b[c] = *(const v2f*)(&ldsB[cur][wn * 32 + c * 16 + lan][kk]);
#pragma unroll
            for (int r = 0; r < 4; ++r)
#pragma unroll
                for (int c = 0; c < 2; ++c)
                    acc[r][c] = __builtin_amdgcn_wmma_f32_16x16x4_f32(
                        false, a[r], false, b[c], (short)0, acc[r][c], false, false);
        }
        __syncthreads();
    }

    // epilogue: d = |z|^2 + |e|^2 - 2*dot ; stream out ; per-row argmin via u64 atomicMin
    float enreg[2];
#pragma unroll
    for (int c = 0; c < 2; ++c) enreg[c] = en[j0 + wn * 32 + c * 16 + lan];

#pragma unroll
    for (int r = 0; r < 4; ++r) {
#pragma unroll
        for (int v = 0; v < 8; ++v) {
            const int gi   = i0 + wm * 64 + r * 16 + v + 8 * hi;  // C layout: VGPR v -> M=v / v+8
            const float zv = zn[gi];
            float best = 3.402823466e38f;
            int bestj  = 0x7fffffff;
#pragma unroll
            for (int c = 0; c < 2; ++c) {
                const int gj   = j0 + wn * 32 + c * 16 + lan;     // C layout: N = lane&15
                const float dv = zv + enreg[c] - 2.0f * acc[r][c][v];
                dOut[(size_t)gi * NE + gj] = dv;
                if (dv < best || (dv == best && gj < bestj)) { best = dv; bestj = gj; }
            }
#pragma unroll
            for (int m = 1; m <= 8; m <<= 1) {   // reduce across the 16 lanes sharing this row
                float ov = __shfl_xor(best, m, 32);
                int   oj = __shfl_xor(bestj, m, 32);
                if (ov < best || (ov == best && oj < bestj)) { best = ov; bestj = oj; }
            }
            if (lan == 0) {
                unsigned long long pack =
                    ((unsigned long long)fkey(best) << 32) | (unsigned int)bestj;
                atomicMin(&minpack[gi], pack);
            }
        }
    }
}

// ---------------------------------------------------------------------------
// one wave per row: gather winning code, write x_q_ste & index, loss partials
__global__ void vq_gather(const float* __restrict__ x, const float* __restrict__ emb,
                          const unsigned long long* __restrict__ minpack,
                          float* __restrict__ xq_out, float* __restrict__ idx_out,
                          float* __restrict__ partials) {
    __shared__ float wsum[8];
    const int wave = threadIdx.x >> 5;
    const int lane = threadIdx.x & 31;
    const int row  = blockIdx.x * 8 + wave;

    const unsigned long long pack = minpack[row];
    const int j = (int)(unsigned int)(pack & 0xFFFFFFFFu);

    const float* e  = emb + (size_t)j * DDIM;
    const float* xr = x + (size_t)row * DDIM;
    float4 e0 = *(const float4*)(e + lane * 4);
    float4 e1 = *(const float4*)(e + 128 + lane * 4);
    float4 x0 = *(const float4*)(xr + lane * 4);
    float4 x1 = *(const float4*)(xr + 128 + lane * 4);
    *(float4*)(xq_out + (size_t)row * DDIM + lane * 4) = e0;
    *(float4*)(xq_out + (size_t)row * DDIM + 128 + lane * 4) = e1;

    float dx, s = 0.f;
    dx = e0.x - x0.x; s += dx * dx;  dx = e0.y - x0.y; s += dx * dx;
    dx = e0.z - x0.z; s += dx * dx;  dx = e0.w - x0.w; s += dx * dx;
    dx = e1.x - x1.x; s += dx * dx;  dx = e1.y - x1.y; s += dx * dx;
    dx = e1.z - x1.z; s += dx * dx;  dx = e1.w - x1.w; s += dx * dx;
#pragma unroll
    for (int m = 16; m >= 1; m >>= 1) s += __shfl_xor(s, m, 32);
    if (lane == 0) {
        wsum[wave]   = s;
        idx_out[row] = (float)j;   // jnp tuple concat promotes indices to f32
    }
    __syncthreads();
    if (threadIdx.x == 0) {
        float t = 0.f;
#pragma unroll
        for (int w = 0; w < 8; ++w) t += wsum[w];
        partials[blockIdx.x] = t;
    }
}

// ---------------------------------------------------------------------------
// deterministic fixed-order final reduction of 8192 partials
__global__ void vq_loss(const float* __restrict__ partials, float* __restrict__ loss_out) {
    __shared__ float sm[256];
    float s = 0.f;
    for (int i = threadIdx.x; i < 8192; i += 256) s += partials[i];
    sm[threadIdx.x] = s;
    __syncthreads();
    for (int off = 128; off > 0; off >>= 1) {
        if ((int)threadIdx.x < off) sm[threadIdx.x] += sm[threadIdx.x + off];
        __syncthreads();
    }
    if (threadIdx.x == 0)
        loss_out[0] = sm[0] * (1.0f + BETA) / 16777216.0f;  // (1+beta)*mean
}

// ---------------------------------------------------------------------------
extern "C" void kernel_launch(void* const* d_in, const int* in_sizes, int n_in,
                              void* d_out, int out_size, void* d_ws, size_t ws_size,
                              hipStream_t stream) {
    const float* x   = (const float*)d_in[0];   // [64,1024,256]
    const float* emb = (const float*)d_in[1];   // [8192,256]
    // d_in[2] = epoch_idx (unused: plain VQ loss path)

    float* out  = (float*)d_out;
    float* xq   = out;                              // 16777216 floats
    float* loss = out + 16777216;                   // 1
    float* idxo = out + 16777217;                   // 65536
    float* dmat = out + 16842753;                   // 65536*8192

    char* ws = (char*)d_ws;
    float*              zn       = (float*)(ws);                         // 65536 f
    float*              en       = (float*)(ws + 262144);                // 8192 f
    unsigned long long* minpack  = (unsigned long long*)(ws + 294912);   // 65536 u64
    float*              partials = (float*)(ws + 819200);                // 8192 f

    vq_init<<<N_ROWS / 256, 256, 0, stream>>>(minpack);
    vq_norms<<<(N_ROWS + NE) / 8, 256, 0, stream>>>(x, emb, zn, en);
    vq_gemm<<<dim3(NE / 128, N_ROWS / 128), 256, 0, stream>>>(x, emb, zn, en, dmat, minpack);
    vq_gather<<<N_ROWS / 8, 256, 0, stream>>>(x, emb, minpack, xq, idxo, partials);
    vq_loss<<<1, 256, 0, stream>>>(partials, loss);
}